// SimpleDeepModel_50697793962023
// MI455X (gfx1250) — compile-verified
//
#include <hip/hip_runtime.h>
#include <hip/hip_bf16.h>
#include <math.h>

typedef float v2f __attribute__((ext_vector_type(2)));
typedef float v4f __attribute__((ext_vector_type(4)));
typedef float v8f __attribute__((ext_vector_type(8)));

#define HIDN 500
#define GATES 2000   // 4*HIDN
#define SEQLEN 4096
#define NCLASS 20

// ---------------------------------------------------------------------------
// Prep: transpose W[N][K] (row-major) -> Wt[K][N] so recurrent/matmul B-loads
// are coalesced.  One-time cost, negligible.
// ---------------------------------------------------------------------------
__global__ void transpose_k(const float* __restrict__ src, float* __restrict__ dst,
                            int N, int K) {
    long i = (long)blockIdx.x * blockDim.x + threadIdx.x;
    if (i < (long)N * K) {
        int n = (int)(i / K);
        int k = (int)(i % K);
        dst[(long)k * N + n] = src[i];
    }
}

__global__ void addbias_k(const float* __restrict__ a, const float* __restrict__ b,
                          float* __restrict__ o, int n) {
    int i = blockIdx.x * blockDim.x + threadIdx.x;
    if (i < n) o[i] = a[i] + b[i];
}

// ---------------------------------------------------------------------------
// WMMA fp32 GEMM:  C[M][N] = A[M][K] @ Bt[K][N] + bias[N]
// One 16x16 output tile per wave32, K stepped by 4 with V_WMMA_F32_16X16X4_F32
// (confirmed lowering in round-1 disasm).  Optional row-gather (idx != nullptr)
// fuses the embedding lookup into the layer-0 projection.  EXEC stays all-ones
// inside the wave (uniform branch).
// ---------------------------------------------------------------------------
__global__ __launch_bounds__(256) void gemm_wmma_f32(
    const float* __restrict__ A, const int* __restrict__ idx, int lda,
    const float* __restrict__ Bt, const float* __restrict__ bias,
    float* __restrict__ C, int M, int N, int K) {
    const int wave       = blockIdx.x * (blockDim.x >> 5) + (threadIdx.x >> 5);
    const int nTilesN    = N >> 4;
    const int totalTiles = (M >> 4) * nTilesN;
    if (wave >= totalTiles) return;                 // wave-uniform: EXEC stays full

    const int mT   = wave / nTilesN;
    const int nT   = wave % nTilesN;
    const int lane = threadIdx.x & 31;
    const int l    = lane & 15;
    const int half = lane >> 4;                     // 0: K=0,1  1: K=2,3
    const int m    = (mT << 4) + l;
    const int n0   = nT << 4;

    const long arow = idx ? (long)idx[m] * lda : (long)m * lda;
    const float* Ap = A + arow + (half << 1);
    const float* Bp = Bt + (long)(half << 1) * N + n0 + l;

    v8f acc = {};
    for (int k = 0; k < K; k += 4) {
        v2f a = *(const v2f*)(Ap + k);              // A[m][k+2h], A[m][k+2h+1]
        v2f b;
        b.x = Bp[(long)k * N];                      // Bt[k+2h  ][n0+l]
        b.y = Bp[(long)(k + 1) * N];                // Bt[k+2h+1][n0+l]
        acc = __builtin_amdgcn_wmma_f32_16x16x4_f32(
            /*neg_a=*/false, a, /*neg_b=*/false, b,
            /*c_mod=*/(short)0, acc, /*reuse_a=*/false, /*reuse_b=*/false);
    }

    const float bn = bias ? bias[n0 + l] : 0.0f;
    float* Cp = C + (long)((mT << 4) + (half << 3)) * N + n0 + l;
#pragma unroll
    for (int r = 0; r < 8; ++r)
        Cp[(long)r * N] = acc[r] + bn;
}

// ---------------------------------------------------------------------------
// Sequential LSTM layer: one persistent 512-thread workgroup (16 wave32s on a
// WGP).  h/c/gates live in LDS (12 KB).  Whht is [K=500][2000] so each k-step
// is one coalesced global_load_b128 of 4 gate weights per thread (L2-resident,
// 4 MB) against a same-address ds_load_b128 broadcast of h[k..k+3]; each
// thread owns gates 4t..4t+3 and accumulates two float2 pairs so the packed
// fp32 pipe (v_pk_fma_f32) carries the math.  Gate order i,f,g,o per
// reference; x_proj already contains (b_ih + b_hh).
// ---------------------------------------------------------------------------
__global__ __launch_bounds__(512) void lstm_seq(
    const float* __restrict__ xproj,   // [S][2000]
    const float* __restrict__ Whht,    // [500][2000]
    float* __restrict__ hout,          // [S][500] or nullptr
    float* __restrict__ hlast,         // [500] or nullptr
    int S) {
    __shared__ float h[HIDN];
    __shared__ float c[HIDN];
    __shared__ float gates[GATES];

    const int tid = threadIdx.x;
    if (tid < HIDN) { h[tid] = 0.0f; c[tid] = 0.0f; }
    __syncthreads();

    const int j0     = tid << 2;                    // 4 gates owned by thread
    const bool gactive = (j0 < GATES);              // threads 0..499

    for (int t = 0; t < S; ++t) {
        if (gactive) {
            const v4f xp = *(const v4f*)(xproj + (long)t * GATES + j0);
            v2f acc01 = {xp.x, xp.y};
            v2f acc23 = {xp.z, xp.w};
            const float* wp = Whht + j0;
#pragma unroll 4
            for (int k = 0; k < HIDN; k += 4) {
                v4f hv = *(const v4f*)(h + k);      // LDS broadcast, 1x b128 / 4k
#pragma unroll
                for (int u = 0; u < 4; ++u) {
                    v4f w = *(const v4f*)(wp + (long)(k + u) * GATES);
                    float s = hv[u];
                    v2f sv = {s, s};
                    acc01 += ((v2f){w.x, w.y}) * sv;   // -> v_pk_fma_f32
                    acc23 += ((v2f){w.z, w.w}) * sv;   // -> v_pk_fma_f32
                }
            }
            v4f g = {acc01.x, acc01.y, acc23.x, acc23.y};
            *(v4f*)(gates + j0) = g;
        }
        __syncthreads();

        if (tid < HIDN) {
            float ig = 1.0f / (1.0f + __expf(-gates[tid]));
            float fg = 1.0f / (1.0f + __expf(-gates[HIDN + tid]));
            float gg = tanhf(gates[2 * HIDN + tid]);
            float og = 1.0f / (1.0f + __expf(-gates[3 * HIDN + tid]));
            float cn = fg * c[tid] + ig * gg;
            c[tid] = cn;
            float hn = og * tanhf(cn);
            h[tid] = hn;
            if (hout)  hout[(long)t * HIDN + tid] = hn;
            if (hlast && t == S - 1) hlast[tid] = hn;
        }
        __syncthreads();
    }
}

// ---------------------------------------------------------------------------
// Final FC: out[20] = h @ fc_W.T + fc_b  (trivial)
// ---------------------------------------------------------------------------
__global__ void fc_k(const float* __restrict__ h, const float* __restrict__ W,
                     const float* __restrict__ b, float* __restrict__ out) {
    int n = threadIdx.x;
    if (n < NCLASS) {
        float a = b[n];
        const float* w = W + n * HIDN;
        for (int k = 0; k < HIDN; ++k) a = fmaf(w[k], h[k], a);
        out[n] = a;
    }
}

// ---------------------------------------------------------------------------
extern "C" void kernel_launch(void* const* d_in, const int* in_sizes, int n_in,
                              void* d_out, int out_size, void* d_ws, size_t ws_size,
                              hipStream_t stream) {
    const int*   seq   = (const int*)  d_in[0];
    const float* emb   = (const float*)d_in[1];
    const float* W_ih0 = (const float*)d_in[2];
    const float* W_hh0 = (const float*)d_in[3];
    const float* b_ih0 = (const float*)d_in[4];
    const float* b_hh0 = (const float*)d_in[5];
    const float* W_ih1 = (const float*)d_in[6];
    const float* W_hh1 = (const float*)d_in[7];
    const float* b_ih1 = (const float*)d_in[8];
    const float* b_hh1 = (const float*)d_in[9];
    const float* fc_W  = (const float*)d_in[10];
    const float* fc_b  = (const float*)d_in[11];
    float*       out   = (float*)d_out;

    float* ws = (float*)d_ws;
    size_t off = 0;
    auto alloc = [&](size_t n) { size_t r = off; off += (n + 63) & ~(size_t)63; return r; };
    float* Wt0   = ws + alloc(300 * 2000);           // W_ih0^T  [300][2000]
    float* Whh0t = ws + alloc(500 * 2000);           // W_hh0^T  [500][2000]
    float* Wt1   = ws + alloc(500 * 2000);           // W_ih1^T  [500][2000]
    float* Whh1t = ws + alloc(500 * 2000);           // W_hh1^T  [500][2000]
    float* bias0 = ws + alloc(2000);
    float* bias1 = ws + alloc(2000);
    float* xproj = ws + alloc((size_t)SEQLEN * GATES);  // reused by both layers
    float* h1    = ws + alloc((size_t)SEQLEN * HIDN);
    float* hlast = ws + alloc(HIDN);

    // --- prep: transposes + fused biases ---
    transpose_k<<<(2000 * 300 + 255) / 256, 256, 0, stream>>>(W_ih0, Wt0,   2000, 300);
    transpose_k<<<(2000 * 500 + 255) / 256, 256, 0, stream>>>(W_hh0, Whh0t, 2000, 500);
    transpose_k<<<(2000 * 500 + 255) / 256, 256, 0, stream>>>(W_ih1, Wt1,   2000, 500);
    transpose_k<<<(2000 * 500 + 255) / 256, 256, 0, stream>>>(W_hh1, Whh1t, 2000, 500);
    addbias_k<<<(2000 + 255) / 256, 256, 0, stream>>>(b_ih0, b_hh0, bias0, 2000);
    addbias_k<<<(2000 + 255) / 256, 256, 0, stream>>>(b_ih1, b_hh1, bias1, 2000);

    // --- layer 0: x_proj GEMM (fused embedding gather) + recurrence ---
    const int tiles = (SEQLEN / 16) * (GATES / 16);  // 256 * 125 = 32000
    const int gemmBlocks = (tiles + 7) / 8;          // 8 waves per 256-thread block
    gemm_wmma_f32<<<gemmBlocks, 256, 0, stream>>>(emb, seq, 300, Wt0, bias0,
                                                  xproj, SEQLEN, GATES, 300);
    lstm_seq<<<1, 512, 0, stream>>>(xproj, Whh0t, h1, nullptr, SEQLEN);

    // --- layer 1: x_proj GEMM on h1 + recurrence (keep only final h) ---
    gemm_wmma_f32<<<gemmBlocks, 256, 0, stream>>>(h1, nullptr, 500, Wt1, bias1,
                                                  xproj, SEQLEN, GATES, 500);
    lstm_seq<<<1, 512, 0, stream>>>(xproj, Whh1t, nullptr, hlast, SEQLEN);

    // --- classifier head ---
    fc_k<<<1, 32, 0, stream>>>(hlast, fc_W, fc_b, out);
    (void)in_sizes; (void)n_in; (void)out_size; (void)ws_size;
}